// SuperVoxelLoss_12953621365099
// MI455X (gfx1250) — compile-verified
//
#include <hip/hip_runtime.h>

// SuperVoxelLoss: weighted BCE-with-logits -> scalar mean.
// Memory-bound streaming reduction: ~134 MB traffic, ~5.8 us floor @ 23.3 TB/s
// (inputs fit in the 192 MB L2, so graph replays run out of L2 with default RT
// temporal hints). Cross-lane reduction uses V_WMMA_F32_16X16X4_F32 (f32, RNE).

typedef __attribute__((ext_vector_type(4))) float v4f;
typedef __attribute__((ext_vector_type(2))) float v2f;
typedef __attribute__((ext_vector_type(8))) float v8f;

#define SVL_ALPHA 0.5f
#define SVL_BETA  0.5f

constexpr int BLOCK  = 256;   // 8 wave32s
constexpr int GRID   = 2048;  // 8 float4 iters/thread, perfectly uniform
constexpr int NWAVES = BLOCK / 32;

__device__ __forceinline__ float svl_elem(float x, float t) {
    // BCEWithLogits (stable): max(x,0) - x*t + log(1 + exp(-|x|)).
    // exp(-|x|) in (0,1] so plain log(1+e) is stable.
    float ax = fabsf(x);
    float l  = fmaxf(x, 0.0f) - x * t + __logf(1.0f + __expf(-ax));
    // pred_bin = sigmoid(x) > 0.5  <=>  x > 0
    float neg = ((t > 0.5f) && (x <= 0.0f)) ? 1.0f : 0.0f;  // false negative
    float pos = ((x > 0.0f) && (t <= 0.5f)) ? 1.0f : 0.0f;  // false positive
    float w = 1.0f + SVL_ALPHA * neg + SVL_BETA * pos;
    return l * w;
}

// Wave32 reduction via one f32 WMMA. Requires EXEC == all ones.
// A(16x4): vgpr0 = psum (lanes 0-15 -> K=0 rows 0-15, lanes 16-31 -> K=2),
//          vgpr1 = 0. B(4x16) = ones => D[m][n] = psum[m] + psum[m+16].
// C/D layout: VGPR j, lanes 0-15 hold D[j][lane]; lanes 16-31 hold D[j+8][.].
// => lane 0's 8-reg sum = sum of rows 0..7, lane 16's = sum of rows 8..15;
// those two together are the full 32-lane sum. Both write to LDS (no shuffle).
__device__ __forceinline__ float wave_half_sum_wmma(float x) {
    v2f a; a[0] = x;    a[1] = 0.0f;
    v2f b; b[0] = 1.0f; b[1] = 1.0f;
    v8f c = {};
    c = __builtin_amdgcn_wmma_f32_16x16x4_f32(
            /*neg_a=*/false, a, /*neg_b=*/false, b,
            /*c_mod=*/(short)0, c, /*reuse_a=*/false, /*reuse_b=*/false);
    return ((c[0] + c[1]) + (c[2] + c[3])) + ((c[4] + c[5]) + (c[6] + c[7]));
}

// Block-level sum; result valid in thread 0 only. Call fully converged.
__device__ __forceinline__ float block_sum(float acc, float* lds) {
    float h = wave_half_sum_wmma(acc);
    // Each wave: lane 0 and lane 16 each hold half the wave sum.
    if ((threadIdx.x & 15) == 0)
        lds[threadIdx.x >> 4] = h;      // 2*NWAVES = 16 entries
    __syncthreads();
    float s = 0.0f;
    if (threadIdx.x == 0) {
#pragma unroll
        for (int i = 0; i < 2 * NWAVES; ++i) s += lds[i];
    }
    return s;
}

__global__ void __launch_bounds__(BLOCK)
svl_partial_kernel(const float* __restrict__ pred,
                   const float* __restrict__ tgt,
                   float* __restrict__ partial,
                   long long n4) {
    __shared__ float lds[2 * NWAVES];
    const v4f* __restrict__ p4 = (const v4f*)pred;
    const v4f* __restrict__ t4 = (const v4f*)tgt;

    long long tid    = (long long)blockIdx.x * BLOCK + threadIdx.x;
    long long stride = (long long)GRID * BLOCK;

    // Two accumulators break the add dependency chain; unroll for MLP.
    float acc0 = 0.0f, acc1 = 0.0f;
#pragma unroll 4
    for (long long i = tid; i < n4; i += stride) {
        v4f x = p4[i];
        v4f t = t4[i];
        acc0 += svl_elem(x[0], t[0]) + svl_elem(x[1], t[1]);
        acc1 += svl_elem(x[2], t[2]) + svl_elem(x[3], t[3]);
    }

    float s = block_sum(acc0 + acc1, lds);  // uniform loop => EXEC all-ones
    if (threadIdx.x == 0) partial[blockIdx.x] = s;
}

__global__ void __launch_bounds__(BLOCK)
svl_final_kernel(const float* __restrict__ partial,
                 float* __restrict__ out, float inv_n) {
    __shared__ float lds[2 * NWAVES];
    const v4f* __restrict__ p4 = (const v4f*)partial;

    // GRID = 2048 partials = 512 float4s = exactly 2 per thread.
    float acc = 0.0f;
#pragma unroll
    for (int j = 0; j < GRID / 4 / BLOCK; ++j) {
        v4f v = p4[j * BLOCK + threadIdx.x];
        acc += (v[0] + v[1]) + (v[2] + v[3]);
    }

    float s = block_sum(acc, lds);
    if (threadIdx.x == 0) out[0] = s * inv_n;
}

extern "C" void kernel_launch(void* const* d_in, const int* in_sizes, int n_in,
                              void* d_out, int out_size, void* d_ws, size_t ws_size,
                              hipStream_t stream) {
    const float* pred = (const float*)d_in[0];
    const float* tgt  = (const float*)d_in[1];
    long long n  = (long long)in_sizes[0];   // 16,777,216 (divisible by 4)
    long long n4 = n >> 2;

    float* partial = (float*)d_ws;           // GRID floats of scratch

    svl_partial_kernel<<<GRID, BLOCK, 0, stream>>>(pred, tgt, partial, n4);
    svl_final_kernel<<<1, BLOCK, 0, stream>>>(partial, (float*)d_out,
                                              1.0f / (float)n);
}